// GraphConvLayer_37280316129401
// MI455X (gfx1250) — compile-verified
//
#include <hip/hip_runtime.h>

typedef __attribute__((ext_vector_type(2))) float v2f;
typedef __attribute__((ext_vector_type(8))) float v8f;

#define N_DIM 1024
#define F_DIM 128

// ---------------------------------------------------------------------------
// K1: Y = U^T @ x fused with diagonal scaling.
//   A[m,k] = U[k,m] (16x4 f32 fragment), B[k,f] = x[k,f] (4x16 fragment)
//   Epilogue: Zt[m*128 + f] = filters[f][m][m] * Y[m,f]   (m-major for K2)
// One wave -> 16(m) x 64(f) output block: 4 WMMAs per K-step sharing one A.
// 128 waves total = 16 blocks x 256 threads. EXEC all ones (no divergence).
// ---------------------------------------------------------------------------
__global__ __launch_bounds__(256)
void gcl_k1_utx_diag(const float* __restrict__ U, const float* __restrict__ x,
                     const float* __restrict__ filters, float* __restrict__ Zt) {
  const int wave = (blockIdx.x * 256 + threadIdx.x) >> 5;  // 0..127
  const int lane = threadIdx.x & 31;
  const int mt   = wave >> 1;           // 0..63  -> m tile
  const int fg   = wave & 1;            // 0..1   -> f group of 64
  const int m0   = mt * 16;
  const int f0   = fg * 64;
  const int lm   = lane & 15;           // M (A) / N (B,C) lane coordinate
  const int kb   = (lane >> 4) * 2;     // K sub-offset: lanes 16-31 hold K=2,3

  v8f acc0 = {}, acc1 = {}, acc2 = {}, acc3 = {};

  for (int k0 = 0; k0 < N_DIM; k0 += 4) {
    const int k = k0 + kb;
    v2f a;
    a.x = U[(size_t)k       * N_DIM + (m0 + lm)];   // A[m0+lm, k]   = U[k, m]
    a.y = U[(size_t)(k + 1) * N_DIM + (m0 + lm)];   // A[m0+lm, k+1]

    const float* xr0 = x + (size_t)k       * F_DIM + f0 + lm;
    const float* xr1 = x + (size_t)(k + 1) * F_DIM + f0 + lm;
    v2f b0, b1, b2, b3;
    b0.x = xr0[0];  b0.y = xr1[0];
    b1.x = xr0[16]; b1.y = xr1[16];
    b2.x = xr0[32]; b2.y = xr1[32];
    b3.x = xr0[48]; b3.y = xr1[48];

    acc0 = __builtin_amdgcn_wmma_f32_16x16x4_f32(false, a, false, b0, (short)0, acc0, false, false);
    acc1 = __builtin_amdgcn_wmma_f32_16x16x4_f32(false, a, false, b1, (short)0, acc1, false, false);
    acc2 = __builtin_amdgcn_wmma_f32_16x16x4_f32(false, a, false, b2, (short)0, acc2, false, false);
    acc3 = __builtin_amdgcn_wmma_f32_16x16x4_f32(false, a, false, b3, (short)0, acc3, false, false);
  }

  // C/D layout: VGPR v, lanes 0-15 -> M=v, lanes 16-31 -> M=v+8; N = lane&15.
  const size_t NN    = (size_t)N_DIM * N_DIM;
  const int    mbase = (lane >> 4) * 8;
  const int    fA    = f0 + lm;
#pragma unroll
  for (int v = 0; v < 8; ++v) {
    const int    m  = m0 + mbase + v;
    const size_t md = (size_t)m * (N_DIM + 1);  // filters[f][m][m] offset within one filter
    Zt[m * F_DIM + fA     ] = filters[(size_t)(fA     ) * NN + md] * acc0[v];
    Zt[m * F_DIM + fA + 16] = filters[(size_t)(fA + 16) * NN + md] * acc1[v];
    Zt[m * F_DIM + fA + 32] = filters[(size_t)(fA + 32) * NN + md] * acc2[v];
    Zt[m * F_DIM + fA + 48] = filters[(size_t)(fA + 48) * NN + md] * acc3[v];
  }
}

// ---------------------------------------------------------------------------
// K2: out[f,n] = sum_m Zt[m*128+f] * U[n,m]   (OUT = Z @ U^T)
//   A[f,m] = Zt[m*128+f]  -> coalesced global loads (lanes sweep f)
//   B[m,n] = U[n,m]       -> staged through LDS: coalesced row reads of U,
//                            transposed store into padded Bs[k][65] (no bank conflicts)
// Block = 128 threads (4 waves), computes 64(f) x 64(n); each wave one f-tile,
// sharing the LDS B panel across its 4 n-tiles. Grid = (N/64=16, F/64=2).
// ---------------------------------------------------------------------------
__global__ __launch_bounds__(128)
void gcl_k2_z_ut(const float* __restrict__ Zt, const float* __restrict__ U,
                 float* __restrict__ out) {
  __shared__ float Bs[32][65];  // 32 k-values x 64 n-values, padded row

  const int t    = threadIdx.x;
  const int lane = t & 31;
  const int wave = t >> 5;                       // 0..3
  const int f0   = blockIdx.y * 64 + wave * 16;  // wave's f tile (WMMA M)
  const int n0   = blockIdx.x * 64;              // block's n range (WMMA N)
  const int lm   = lane & 15;
  const int kb   = (lane >> 4) * 2;

  v8f acc0 = {}, acc1 = {}, acc2 = {}, acc3 = {};

  for (int kc = 0; kc < N_DIM; kc += 32) {
    // Stage B panel: Bs[k][n] = U[n0+n, kc+k]; 2048 elems / 128 threads = 16 each.
#pragma unroll
    for (int i = 0; i < 16; ++i) {
      const int idx = t + i * 128;   // 0..2047
      const int n   = idx >> 5;      // 0..63
      const int k   = idx & 31;      // 0..31
      Bs[k][n] = U[(size_t)(n0 + n) * N_DIM + kc + k];
    }
    __syncthreads();

#pragma unroll
    for (int k4 = 0; k4 < 32; k4 += 4) {
      v2f a;
      const float* zp = Zt + (size_t)(kc + k4 + kb) * F_DIM + f0 + lm;
      a.x = zp[0];
      a.y = zp[F_DIM];

      v2f b0, b1, b2, b3;
      b0.x = Bs[k4 + kb][lm     ]; b0.y = Bs[k4 + kb + 1][lm     ];
      b1.x = Bs[k4 + kb][lm + 16]; b1.y = Bs[k4 + kb + 1][lm + 16];
      b2.x = Bs[k4 + kb][lm + 32]; b2.y = Bs[k4 + kb + 1][lm + 32];
      b3.x = Bs[k4 + kb][lm + 48]; b3.y = Bs[k4 + kb + 1][lm + 48];

      acc0 = __builtin_amdgcn_wmma_f32_16x16x4_f32(false, a, false, b0, (short)0, acc0, false, false);
      acc1 = __builtin_amdgcn_wmma_f32_16x16x4_f32(false, a, false, b1, (short)0, acc1, false, false);
      acc2 = __builtin_amdgcn_wmma_f32_16x16x4_f32(false, a, false, b2, (short)0, acc2, false, false);
      acc3 = __builtin_amdgcn_wmma_f32_16x16x4_f32(false, a, false, b3, (short)0, acc3, false, false);
    }
    __syncthreads();
  }

  // Store out (F=128, N=1024) row-major; lanes 0-15 write 64B runs (coalesced).
  const int mbase = (lane >> 4) * 8;
#pragma unroll
  for (int v = 0; v < 8; ++v) {
    const int f = f0 + mbase + v;
    float* o = out + (size_t)f * N_DIM + n0 + lm;
    o[0]  = acc0[v];
    o[16] = acc1[v];
    o[32] = acc2[v];
    o[48] = acc3[v];
  }
}

extern "C" void kernel_launch(void* const* d_in, const int* in_sizes, int n_in,
                              void* d_out, int out_size, void* d_ws, size_t ws_size,
                              hipStream_t stream) {
  (void)in_sizes; (void)n_in; (void)out_size; (void)ws_size;
  const float* x       = (const float*)d_in[0];  // (1024, 128)
  const float* U       = (const float*)d_in[1];  // (1024, 1024)
  const float* filters = (const float*)d_in[2];  // (128, 1024, 1024), diagonal
  float*       out     = (float*)d_out;          // (128, 1024)
  float*       Zt      = (float*)d_ws;           // (1024, 128) scratch, 512 KB

  gcl_k1_utx_diag<<<dim3(16), 256, 0, stream>>>(U, x, filters, Zt);
  gcl_k2_z_ut<<<dim3(16, 2), 128, 0, stream>>>(Zt, U, out);
}